// SVAE_LDS_65601330479123
// MI455X (gfx1250) — compile-verified
//
#include <hip/hip_runtime.h>
#include <hip/hip_bf16.h>

// CDNA5 / gfx1250. wave32. WMMA f32 16x16x4 for the tiny-K dense GEMMs,
// lane-distributed 4x4 algebra for the sequential Kalman scan.
//
// Roofline: parallel phases move ~120MB total -> ~5us at 23.3 TB/s; the
// sequential 400k-step scan is latency-bound, so its loop body is built to
// minimize dependent cross-lane hops and to keep stores off the chain.

typedef float v2f __attribute__((ext_vector_type(2)));
typedef float v8f __attribute__((ext_vector_type(8)));

#define D4 4
#define OBS 10
#define HID 6

// ---------------------------------------------------------------------------
// helpers
// ---------------------------------------------------------------------------
__device__ __forceinline__ float sel4(int s, float a, float b, float c, float d) {
    return (s == 0) ? a : (s == 1) ? b : (s == 2) ? c : d;
}

__device__ __forceinline__ unsigned hash32(unsigned x) {
    x ^= x >> 16; x *= 0x7feb352du;
    x ^= x >> 15; x *= 0x846ca68bu;
    x ^= x >> 16; return x;
}

// Cholesky of a 4x4 SPD matrix held fully (row-major) per lane.
__device__ __forceinline__ void chol4(const float* S, float* L) {
    float l00 = sqrtf(fmaxf(S[0], 1e-30f));            float i00 = 1.f / l00;
    float l10 = S[4]  * i00, l20 = S[8] * i00, l30 = S[12] * i00;
    float l11 = sqrtf(fmaxf(S[5] - l10 * l10, 1e-30f)); float i11 = 1.f / l11;
    float l21 = (S[9]  - l20 * l10) * i11;
    float l31 = (S[13] - l30 * l10) * i11;
    float l22 = sqrtf(fmaxf(S[10] - l20 * l20 - l21 * l21, 1e-30f)); float i22 = 1.f / l22;
    float l32 = (S[14] - l30 * l20 - l31 * l21) * i22;
    float l33 = sqrtf(fmaxf(S[15] - l30 * l30 - l31 * l31 - l32 * l32, 1e-30f));
    L[0] = l00; L[1] = 0;   L[2] = 0;   L[3] = 0;
    L[4] = l10; L[5] = l11; L[6] = 0;   L[7] = 0;
    L[8] = l20; L[9] = l21; L[10] = l22; L[11] = 0;
    L[12] = l30; L[13] = l31; L[14] = l32; L[15] = l33;
}

// zero one 256-float LDS image with 2 x ds_store_b128 per lane (single wave)
__device__ __forceinline__ void lds_zero256(float* p, int l) {
    float4 z = make_float4(0.f, 0.f, 0.f, 0.f);
    ((float4*)p)[l]      = z;
    ((float4*)p)[l + 32] = z;
}

// ---------------------------------------------------------------------------
// Encoder: h = relu(x@W1+b1); zm = h@Wm+bm; sd = exp(0.5*(h@Wv+bv))
// One wave per 16-row tile. Biases are folded in as an extra K-row
// (ones column in A, bias row in B) so all fragment reads are unguarded
// LDS reads of zero-padded images. Stage 1: K = 10 data + 1 bias -> 3 WMMA.
// ---------------------------------------------------------------------------
__global__ __launch_bounds__(32) void enc_kernel(
    const float* __restrict__ x,
    const float* __restrict__ W1, const float* __restrict__ b1,
    const float* __restrict__ Wm, const float* __restrict__ bm,
    const float* __restrict__ Wv, const float* __restrict__ bv,
    float* __restrict__ zm, float* __restrict__ sd, int T)
{
    __shared__ float xs[16 * 16];   // x tile, col 10 = 1.0 (bias), rest 0
    __shared__ float w1s[16 * 16];  // W1 rows 0..9, row 10 = b1, rest 0
    __shared__ float w2s[16 * 16];  // [Wm|Wv] rows 0..5, row 6 = [bm|bv]
    __shared__ float hs[16 * 17];   // h tile (stride 17), col 6 = 1.0

    const int l    = threadIdx.x;
    const int ln15 = l & 15;
    const int half = l >> 4;
    const int row0 = (int)blockIdx.x * 16;

    lds_zero256(xs, l);
    lds_zero256(w1s, l);
    lds_zero256(w2s, l);

    // ---- cooperative, coalesced fills (same-wave LDS ops are in-order) ----
    const int xlim = T * OBS - row0 * OBS;   // valid elements in this tile
    #pragma unroll
    for (int c = 0; c < 5; ++c) {            // 160 contiguous floats
        int f = l + 32 * c;
        if (f < xlim) xs[(f / OBS) * 16 + (f % OBS)] = x[row0 * OBS + f];
    }
    if (l < 16) xs[l * 16 + OBS] = 1.f;      // bias ones-column (k = 10)

    {   // W1 (10x6 = 60) + b1 row
        int f0 = l;        if (f0 < 60) w1s[(f0 / HID) * 16 + (f0 % HID)] = W1[f0];
        int f1 = l + 32;   if (f1 < 60) w1s[(f1 / HID) * 16 + (f1 % HID)] = W1[f1];
        if (l < HID) w1s[OBS * 16 + l] = b1[l];
    }
    {   // [Wm | Wv] (6x4 each) + [bm | bv] row at k = 6
        if (l < 24) w2s[(l / 4) * 16 + (l % 4)]     = Wm[l];
        if (l < 24) w2s[(l / 4) * 16 + 4 + (l % 4)] = Wv[l];
        if (l < 8)  w2s[HID * 16 + l] = (l < 4) ? bm[l] : bv[l - 4];
    }
    __syncthreads();

    // ---- stage 1: C1 = [x|1](16x11) @ [W1;b1](11x6), K chunks 0..2 ----
    v8f c1;
    #pragma unroll
    for (int r = 0; r < 8; ++r) c1[r] = 0.f;
    #pragma unroll
    for (int kc = 0; kc < 3; ++kc) {
        const int kb = kc * 4;
        v2f a, bf;
        #pragma unroll
        for (int r = 0; r < 2; ++r) {
            int k = kb + 2 * half + r;
            a[r]  = xs[ln15 * 16 + k];
            bf[r] = w1s[k * 16 + ln15];
        }
        c1 = __builtin_amdgcn_wmma_f32_16x16x4_f32(false, a, false, bf,
                                                   (short)0, c1, false, false);
    }

    // relu -> LDS re-layout (C-frag -> A-frag), then bias ones-column
    #pragma unroll
    for (int r = 0; r < 8; ++r) {
        int m = r + 8 * half;
        hs[m * 17 + ln15] = fmaxf(c1[r], 0.f);
    }
    if (l < 16) hs[l * 17 + HID] = 1.f;      // k = 6 ones-column
    __syncthreads();

    // ---- stage 2: [zm|logv] = [h|1](16x7) @ [Wmv;b](7x8), K chunks 0..1 ----
    v8f c2;
    #pragma unroll
    for (int r = 0; r < 8; ++r) c2[r] = 0.f;
    #pragma unroll
    for (int kc = 0; kc < 2; ++kc) {
        const int kb = kc * 4;
        v2f a, bf;
        #pragma unroll
        for (int r = 0; r < 2; ++r) {
            int k = kb + 2 * half + r;
            a[r]  = hs[ln15 * 17 + k];
            bf[r] = w2s[k * 16 + ln15];
        }
        c2 = __builtin_amdgcn_wmma_f32_16x16x4_f32(false, a, false, bf,
                                                   (short)0, c2, false, false);
    }

    // extraction: cols 0..3 -> zm, cols 4..7 -> sd (single guard)
    const int col   = ln15 & 3;
    const bool isv  = ln15 >= 4;
    float* outbase  = isv ? sd : zm;
    #pragma unroll
    for (int r = 0; r < 8; ++r) {
        int grow = row0 + r + 8 * half;
        float v  = c2[r];
        float ov = isv ? __expf(0.5f * v) : v;
        if (ln15 < 8 && grow < T) outbase[grow * D4 + col] = ov;
    }
}

// ---------------------------------------------------------------------------
// Decoder: x_recon = relu(zhat@dW1+db1) @ dW2 + db2 (bias folded as K-row)
// ---------------------------------------------------------------------------
__global__ __launch_bounds__(32) void dec_kernel(
    const float* __restrict__ zh,
    const float* __restrict__ dW1, const float* __restrict__ db1,
    const float* __restrict__ dW2, const float* __restrict__ db2,
    float* __restrict__ xr, int T)
{
    __shared__ float zs[16 * 16];   // zhat tile, col 4 = 1.0
    __shared__ float w1s[16 * 16];  // dW1 rows 0..3, row 4 = db1
    __shared__ float w2s[16 * 16];  // dW2 rows 0..5, row 6 = db2
    __shared__ float hs[16 * 17];   // h tile, col 6 = 1.0

    const int l    = threadIdx.x;
    const int ln15 = l & 15;
    const int half = l >> 4;
    const int row0 = (int)blockIdx.x * 16;

    lds_zero256(zs, l);
    lds_zero256(w1s, l);
    lds_zero256(w2s, l);

    const int zlim = T * D4 - row0 * D4;
    {   // zhat tile: 64 contiguous floats
        int f0 = l;        if (f0 < zlim && f0 < 64) zs[(f0 / 4) * 16 + (f0 % 4)] = zh[row0 * D4 + f0];
        int f1 = l + 32;   if (f1 < zlim && f1 < 64) zs[(f1 / 4) * 16 + (f1 % 4)] = zh[row0 * D4 + f1];
        if (l < 16) zs[l * 16 + D4] = 1.f;   // ones column (k = 4)
    }
    {   // dW1 (4x6 = 24) + db1 row at k = 4
        if (l < 24) w1s[(l / HID) * 16 + (l % HID)] = dW1[l];
        if (l < HID) w1s[D4 * 16 + l] = db1[l];
    }
    {   // dW2 (6x10 = 60) + db2 row at k = 6
        int f0 = l;        if (f0 < 60) w2s[(f0 / OBS) * 16 + (f0 % OBS)] = dW2[f0];
        int f1 = l + 32;   if (f1 < 60) w2s[(f1 / OBS) * 16 + (f1 % OBS)] = dW2[f1];
        if (l < OBS) w2s[HID * 16 + l] = db2[l];
    }
    __syncthreads();

    // ---- stage 1: h = relu([zhat|1](16x5) @ [dW1;db1](5x6)), 2 WMMA ----
    v8f c1;
    #pragma unroll
    for (int r = 0; r < 8; ++r) c1[r] = 0.f;
    #pragma unroll
    for (int kc = 0; kc < 2; ++kc) {
        const int kb = kc * 4;
        v2f a, bf;
        #pragma unroll
        for (int r = 0; r < 2; ++r) {
            int k = kb + 2 * half + r;
            a[r]  = zs[ln15 * 16 + k];
            bf[r] = w1s[k * 16 + ln15];
        }
        c1 = __builtin_amdgcn_wmma_f32_16x16x4_f32(false, a, false, bf,
                                                   (short)0, c1, false, false);
    }
    #pragma unroll
    for (int r = 0; r < 8; ++r) {
        int m = r + 8 * half;
        hs[m * 17 + ln15] = fmaxf(c1[r], 0.f);
    }
    if (l < 16) hs[l * 17 + HID] = 1.f;
    __syncthreads();

    // ---- stage 2: xr = [h|1](16x7) @ [dW2;db2](7x10), 2 WMMA ----
    v8f c2;
    #pragma unroll
    for (int r = 0; r < 8; ++r) c2[r] = 0.f;
    #pragma unroll
    for (int kc = 0; kc < 2; ++kc) {
        const int kb = kc * 4;
        v2f a, bf;
        #pragma unroll
        for (int r = 0; r < 2; ++r) {
            int k = kb + 2 * half + r;
            a[r]  = hs[ln15 * 17 + k];
            bf[r] = w2s[k * 16 + ln15];
        }
        c2 = __builtin_amdgcn_wmma_f32_16x16x4_f32(false, a, false, bf,
                                                   (short)0, c2, false, false);
    }
    #pragma unroll
    for (int r = 0; r < 8; ++r) {
        int grow = row0 + r + 8 * half;
        if (grow < T && ln15 < OBS)
            __builtin_nontemporal_store(c2[r], xr + grow * OBS + ln15);
    }
}

// ---------------------------------------------------------------------------
// eps ~ N(0,1): counter-based hash + Box-Muller (keeps transcendentals off
// the sequential critical path).
// ---------------------------------------------------------------------------
__global__ void eps_kernel(float* __restrict__ eps, int npairs,
                           const int* __restrict__ seedp)
{
    int p = blockIdx.x * blockDim.x + threadIdx.x;
    if (p >= npairs) return;
    unsigned seed = (unsigned)seedp[0];
    unsigned h1 = hash32(0x9E3779B9u ^ (unsigned)(2 * p)     ^ (seed * 0x85ebca6bu));
    unsigned h2 = hash32(0x1B873593u ^ (unsigned)(2 * p + 1) ^ (seed * 0xc2b2ae35u));
    float u1 = ((h1 >> 9) + 1u) * (1.0f / 8388609.0f);   // (0,1]
    float u2 = (h2 >> 8) * (1.0f / 16777216.0f);         // [0,1)
    float r = sqrtf(-2.f * __logf(u1));
    float s, c;
    __sincosf(6.28318530718f * u2, &s, &c);
    eps[2 * p]     = r * c;
    eps[2 * p + 1] = r * s;
}

// ---------------------------------------------------------------------------
// Sequential Kalman scan: one wave32. Lane 4i+j owns element (i,j) of every
// 4x4 matrix; lanes 16..31 mirror 0..15 so EXEC stays full and stores need
// no guards (mirror lanes write identical data to identical addresses).
// Shared scalars (chol, S^-1, means) are gathered once and computed
// redundantly in all lanes: redundancy instead of serialization.
// ---------------------------------------------------------------------------
__global__ __launch_bounds__(32) void scan_kernel(
    const float* __restrict__ zm, const float* __restrict__ sd,
    const float* __restrict__ eps,
    const float* __restrict__ Ain, const float* __restrict__ bin,
    const float* __restrict__ qp,
    float* __restrict__ q_mean, float* __restrict__ q_cov,
    float* __restrict__ p_mean, float* __restrict__ p_cov,
    float* __restrict__ zhat_out, int T)
{
    const int l = threadIdx.x;
    const int i = (l >> 2) & 3;
    const int j = l & 3;

    // uniform constants
    float A[16], b[4], Q[16];
    #pragma unroll
    for (int k = 0; k < 16; ++k) A[k] = Ain[k];
    #pragma unroll
    for (int k = 0; k < 4; ++k) b[k] = bin[k];
    {
        // tfp fill_triangular: arr = concat(qp[4:10], reverse(qp[0:10]))
        float qv[10];
        #pragma unroll
        for (int k = 0; k < 10; ++k) qv[k] = qp[k];
        float arr[16];
        #pragma unroll
        for (int k = 0; k < 6; ++k)  arr[k] = qv[4 + k];
        #pragma unroll
        for (int k = 0; k < 10; ++k) arr[6 + k] = qv[9 - k];
        float L[16];
        #pragma unroll
        for (int r = 0; r < 4; ++r)
            #pragma unroll
            for (int c = 0; c < 4; ++c)
                L[r * 4 + c] = (r > c) ? arr[r * 4 + c]
                             : (r == c) ? __expf(arr[r * 4 + c]) : 0.f;
        #pragma unroll
        for (int r = 0; r < 4; ++r)
            #pragma unroll
            for (int c = 0; c < 4; ++c) {
                float s = 0.f;
                #pragma unroll
                for (int k = 0; k < 4; ++k) s += L[r * 4 + k] * L[c * 4 + k];
                Q[r * 4 + c] = s;
            }
    }

    // carry: mu (uniform), P (distributed element (i,j))
    float mu0 = 0.f, mu1 = 0.f, mu2 = 0.f, mu3 = 0.f;
    float P = (i == j) ? 1.f : 0.f;

    for (int t = 0; t < T; ++t) {
        const float4 zm4 = *(const float4*)(zm + 4 * t);
        const float4 sd4 = *(const float4*)(sd + 4 * t);
        const float4 e4  = *(const float4*)(eps + 4 * t);
        if (t + 256 < T) {   // global_prefetch ahead of the dependent chain
            __builtin_prefetch(zm + 4 * (t + 256), 0, 0);
            __builtin_prefetch(sd + 4 * (t + 256), 0, 0);
            __builtin_prefetch(eps + 4 * (t + 256), 0, 0);
        }

        // P_p = A @ P @ A^T + Q   (distributed, 8 shuffles + 8 FMA)
        float t1 = 0.f;
        #pragma unroll
        for (int k = 0; k < 4; ++k) t1 += A[i * 4 + k] * __shfl(P, 4 * k + j, 32);
        float pp = Q[i * 4 + j];
        #pragma unroll
        for (int k = 0; k < 4; ++k) pp += __shfl(t1, 4 * i + k, 32) * A[j * 4 + k];

        // mu_p = mu @ A + b  (uniform)
        float mp0 = b[0] + mu0 * A[0] + mu1 * A[4] + mu2 * A[8]  + mu3 * A[12];
        float mp1 = b[1] + mu0 * A[1] + mu1 * A[5] + mu2 * A[9]  + mu3 * A[13];
        float mp2 = b[2] + mu0 * A[2] + mu1 * A[6] + mu2 * A[10] + mu3 * A[14];
        float mp3 = b[3] + mu0 * A[3] + mu1 * A[7] + mu2 * A[11] + mu3 * A[15];

        // S = P_p + diag(sd)
        float dg = sel4(i, sd4.x, sd4.y, sd4.z, sd4.w);
        float se = pp + ((i == j) ? dg : 0.f);

        // gather S, invert via Cholesky (redundant in every lane)
        float S[16];
        #pragma unroll
        for (int kk = 0; kk < 16; ++kk) S[kk] = __shfl(se, kk, 32);
        float Ls[16];
        chol4(S, Ls);
        float M00 = 1.f / Ls[0],  M11 = 1.f / Ls[5];
        float M22 = 1.f / Ls[10], M33 = 1.f / Ls[15];
        float M10 = -Ls[4] * M00 * M11;
        float M20 = -(Ls[8] * M00 + Ls[9] * M10) * M22;
        float M21 = -Ls[9] * M11 * M22;
        float M30 = -(Ls[12] * M00 + Ls[13] * M10 + Ls[14] * M20) * M33;
        float M31 = -(Ls[13] * M11 + Ls[14] * M21) * M33;
        float M32 = -Ls[14] * M22 * M33;
        float Si[16];
        Si[0]  = M00 * M00 + M10 * M10 + M20 * M20 + M30 * M30;
        Si[1]  = M10 * M11 + M20 * M21 + M30 * M31;
        Si[2]  = M20 * M22 + M30 * M32;
        Si[3]  = M30 * M33;
        Si[5]  = M11 * M11 + M21 * M21 + M31 * M31;
        Si[6]  = M21 * M22 + M31 * M32;
        Si[7]  = M31 * M33;
        Si[10] = M22 * M22 + M32 * M32;
        Si[11] = M32 * M33;
        Si[15] = M33 * M33;
        Si[4] = Si[1]; Si[8] = Si[2]; Si[12] = Si[3];
        Si[9] = Si[6]; Si[13] = Si[7]; Si[14] = Si[11];

        // K = P_p @ S^-1 (distributed; Si is uniform per-lane)
        float ke = 0.f;
        #pragma unroll
        for (int k = 0; k < 4; ++k)
            ke += __shfl(pp, 4 * i + k, 32)
                * sel4(j, Si[4 * k], Si[4 * k + 1], Si[4 * k + 2], Si[4 * k + 3]);
        float Kf[16];
        #pragma unroll
        for (int kk = 0; kk < 16; ++kk) Kf[kk] = __shfl(ke, kk, 32);

        // mu_u = mu_p + K (zm - mu_p)  (uniform)
        float d0 = zm4.x - mp0, d1 = zm4.y - mp1, d2 = zm4.z - mp2, d3 = zm4.w - mp3;
        float mu_u0 = mp0 + Kf[0]  * d0 + Kf[1]  * d1 + Kf[2]  * d2 + Kf[3]  * d3;
        float mu_u1 = mp1 + Kf[4]  * d0 + Kf[5]  * d1 + Kf[6]  * d2 + Kf[7]  * d3;
        float mu_u2 = mp2 + Kf[8]  * d0 + Kf[9]  * d1 + Kf[10] * d2 + Kf[11] * d3;
        float mu_u3 = mp3 + Kf[12] * d0 + Kf[13] * d1 + Kf[14] * d2 + Kf[15] * d3;

        // P_u = (I - K) P_p  (distributed)
        float pu = pp;
        #pragma unroll
        for (int k = 0; k < 4; ++k)
            pu -= sel4(i, Kf[k], Kf[4 + k], Kf[8 + k], Kf[12 + k])
                * __shfl(pp, 4 * k + j, 32);

        // Pc = 0.5 (P_u + P_u^T); cholesky; z_hat = mu_u + Lc @ e
        float pc = 0.5f * (pu + __shfl(pu, 4 * j + i, 32));
        float Pc[16];
        #pragma unroll
        for (int kk = 0; kk < 16; ++kk) Pc[kk] = __shfl(pc, kk, 32);
        float Lc[16];
        chol4(Pc, Lc);
        float zh0 = mu_u0 + Lc[0]  * e4.x;
        float zh1 = mu_u1 + Lc[4]  * e4.x + Lc[5]  * e4.y;
        float zh2 = mu_u2 + Lc[8]  * e4.x + Lc[9]  * e4.y + Lc[10] * e4.z;
        float zh3 = mu_u3 + Lc[12] * e4.x + Lc[13] * e4.y + Lc[14] * e4.z + Lc[15] * e4.w;

        // stores: all 32 lanes participate (mirror lanes duplicate), no exec
        // churn inside the loop; covariances/means are write-once -> NT.
        __builtin_nontemporal_store(pp, p_cov + t * 16 + (l & 15));
        __builtin_nontemporal_store(pu, q_cov + t * 16 + (l & 15));
        __builtin_nontemporal_store(sel4(j, mp0, mp1, mp2, mp3),     p_mean + t * 4 + j);
        __builtin_nontemporal_store(sel4(j, mu_u0, mu_u1, mu_u2, mu_u3), q_mean + t * 4 + j);
        zhat_out[t * 4 + j] = sel4(j, zh0, zh1, zh2, zh3);  // re-read by decoder

        // carry
        mu0 = mu_u0; mu1 = mu_u1; mu2 = mu_u2; mu3 = mu_u3;
        P = pu;
    }
}

// ---------------------------------------------------------------------------
// launch
// ---------------------------------------------------------------------------
extern "C" void kernel_launch(void* const* d_in, const int* in_sizes, int n_in,
                              void* d_out, int out_size, void* d_ws, size_t ws_size,
                              hipStream_t stream) {
    const float* x      = (const float*)d_in[0];
    const float* enc_W1 = (const float*)d_in[1];
    const float* enc_b1 = (const float*)d_in[2];
    const float* enc_Wm = (const float*)d_in[3];
    const float* enc_bm = (const float*)d_in[4];
    const float* enc_Wv = (const float*)d_in[5];
    const float* enc_bv = (const float*)d_in[6];
    const float* dec_W1 = (const float*)d_in[7];
    const float* dec_b1 = (const float*)d_in[8];
    const float* dec_W2 = (const float*)d_in[9];
    const float* dec_b2 = (const float*)d_in[10];
    const float* A      = (const float*)d_in[11];
    const float* bvec   = (const float*)d_in[12];
    const float* qparam = (const float*)d_in[13];
    const int*   seedp  = (const int*)d_in[14];

    const int T = in_sizes[0] / OBS;

    // workspace: zm | sd | eps | zhat  (each T*4 floats)
    float* w   = (float*)d_ws;
    float* zm  = w;
    float* sd  = w + (size_t)4 * T;
    float* ep  = w + (size_t)8 * T;
    float* zh  = w + (size_t)12 * T;

    // outputs flat, in return order
    float* o      = (float*)d_out;
    float* xr     = o;                       // (T,10)
    float* q_mean = o + (size_t)10 * T;      // (T,4)
    float* q_cov  = o + (size_t)14 * T;      // (T,4,4)
    float* p_mean = o + (size_t)30 * T;      // (T,4)
    float* p_cov  = o + (size_t)34 * T;      // (T,4,4)

    const int ntiles = (T + 15) / 16;
    enc_kernel<<<ntiles, 32, 0, stream>>>(x, enc_W1, enc_b1, enc_Wm, enc_bm,
                                          enc_Wv, enc_bv, zm, sd, T);
    const int npairs = 2 * T;  // T*4 normals, 2 per thread
    eps_kernel<<<(npairs + 255) / 256, 256, 0, stream>>>(ep, npairs, seedp);
    scan_kernel<<<1, 32, 0, stream>>>(zm, sd, ep, A, bvec, qparam,
                                      q_mean, q_cov, p_mean, p_cov, zh, T);
    dec_kernel<<<ntiles, 32, 0, stream>>>(zh, dec_W1, dec_b1, dec_W2, dec_b2, xr, T);
}